// CBA_88854283419703
// MI455X (gfx1250) — compile-verified
//
#include <hip/hip_runtime.h>
#include <hip/hip_bf16.h>

// ---------------------------------------------------------------------------
// Problem constants (reference: B=32, L=2048, E=1024, R=1024, fp32)
// ---------------------------------------------------------------------------
#define B_  32
#define L_  2048
#define E_  1024
#define R_  1024
#define F_  (E_ + R_)       // 2048 features: [prnt_lba(1024) | embs(1024)]
#define S_  16              // L-split factor for the rnn accumulation pass
#define EPS 1e-7f

typedef __attribute__((ext_vector_type(2))) float v2f;
typedef __attribute__((ext_vector_type(4))) float f4;
typedef __attribute__((ext_vector_type(8))) float v8f;

// Workspace layout (in floats). d_ws is assumed >= ~2.3 MB.
#define WS_WSUM  0                         // [2048]   Wsum = W.sum(axis=1)
#define WS_W     (WS_WSUM + F_)            // [B*L]    w_raw = exp(tanh(score))
#define WS_SCALE (WS_W + B_ * L_)          // [B]      1/(sum_l w_raw + eps)
#define WS_PART  (WS_SCALE + 32)           // [S*B*R]  partial weighted sums
// WS_PART float offset = 67616 -> byte offset 270464, 16B aligned. OK for f4.

// ---------------------------------------------------------------------------
// Kernel 1: Wsum[f] = sum_j W[f, j]   (W is [F_, R_] row-major)
// One block per row; 256 threads; LDS tree reduction.
// ---------------------------------------------------------------------------
__global__ void k_wsum(const float* __restrict__ W, float* __restrict__ wsum) {
    __shared__ float red[256];
    const int row = blockIdx.x;                         // 0..F_-1
    const float* wr = W + (size_t)row * R_;
    float s = 0.0f;
    for (int j = threadIdx.x; j < R_; j += 256) s += wr[j];
    red[threadIdx.x] = s;
    __syncthreads();
    for (int off = 128; off > 0; off >>= 1) {
        if (threadIdx.x < off) red[threadIdx.x] += red[threadIdx.x + off];
        __syncthreads();
    }
    if (threadIdx.x == 0) wsum[row] = red[0];
}

// ---------------------------------------------------------------------------
// Kernel 2: per (b,l) score via V_WMMA_F32_16X16X4_F32.
//
// One wave32 owns 16 consecutive rows (row = b*L + l). Per the CDNA5 ISA
// fp32 16x4 A layout: lane m (m<16) holds A[M=m, K={0,1}] in 2 VGPRs,
// lane m+16 holds A[M=m, K={2,3}]. So every lane loads one float2 at
// rowBase + f0 + 2*(lane>>4).  B (4x16) is Wsum broadcast across all 16
// columns: the same per-lane float2 addressing on Wsum reproduces
// B[K=k, N=*] = Wsum[f0+k] under the symmetric B layout.
// After accumulating over K=2048, D column N=0 holds the 16 dot products:
// M=0..7 live in c[0..7] of lane 0; M=8..15 in c[0..7] of lane 16.
// ---------------------------------------------------------------------------
__global__ void k_scores(const float* __restrict__ embs,
                         const long long* __restrict__ prnt,
                         const float* __restrict__ lba,
                         const float* __restrict__ wsum,
                         float* __restrict__ wbuf) {
    const int lane = threadIdx.x & 31;
    const int wave = threadIdx.x >> 5;
    const int row0 = blockIdx.x * 128 + wave * 16;      // first of this wave's 16 rows
    const int m    = lane & 15;
    const int row  = row0 + m;                          // this lane's row = b*L + l
    const int b    = row >> 11;                         // L_ == 2048
    const long long p = prnt[row];                      // parent index in [0, L)
    const float* lbaBase = lba  + (((size_t)b << 11) + (size_t)p) * R_;
    const float* embBase = embs + (size_t)row * E_;
    const int koff = (lane >> 4) << 1;                  // 0 for lanes 0-15, 2 for 16-31

    v8f c = {};
    // Segment 1: gathered parent lba row against Wsum[0:1024)
#pragma unroll 4
    for (int f = 0; f < R_; f += 4) {
        v2f a  = *(const v2f*)(lbaBase + f + koff);
        v2f bb = *(const v2f*)(wsum    + f + koff);
        c = __builtin_amdgcn_wmma_f32_16x16x4_f32(false, a, false, bb,
                                                  (short)0, c, false, false);
    }
    // Segment 2: embs row against Wsum[1024:2048)
#pragma unroll 4
    for (int f = 0; f < E_; f += 4) {
        v2f a  = *(const v2f*)(embBase + f + koff);
        v2f bb = *(const v2f*)(wsum + R_ + f + koff);
        c = __builtin_amdgcn_wmma_f32_16x16x4_f32(false, a, false, bb,
                                                  (short)0, c, false, false);
    }

    // Extract column N=0 and apply exp(tanh(.)).
    if (lane == 0) {
#pragma unroll
        for (int j = 0; j < 8; ++j)
            wbuf[row0 + j] = __expf(tanhf(c[j]));
    } else if (lane == 16) {
#pragma unroll
        for (int j = 0; j < 8; ++j)
            wbuf[row0 + 8 + j] = __expf(tanhf(c[j]));
    }
}

// ---------------------------------------------------------------------------
// Kernel 3: scale[b] = 1 / (sum_l w_raw[b,l] + eps)
// ---------------------------------------------------------------------------
__global__ void k_scale(const float* __restrict__ wbuf, float* __restrict__ scale) {
    __shared__ float red[256];
    const int b = blockIdx.x;
    const float* wb = wbuf + (size_t)b * L_;
    float s = 0.0f;
    for (int l = threadIdx.x; l < L_; l += 256) s += wb[l];
    red[threadIdx.x] = s;
    __syncthreads();
    for (int off = 128; off > 0; off >>= 1) {
        if (threadIdx.x < off) red[threadIdx.x] += red[threadIdx.x + off];
        __syncthreads();
    }
    if (threadIdx.x == 0) scale[b] = 1.0f / (red[0] + EPS);
}

// ---------------------------------------------------------------------------
// Kernel 4: partial[s][b][r] = sum over L-chunk s of w_raw[b,l] * rnn[b,l,r]
// Grid (B, S); 256 threads x float4 covers the full R=1024 per block.
// This is the dominant 256 MB streaming pass: b128 coalesced loads of rnn,
// block-uniform scalar broadcast of w_raw[b,l].
// ---------------------------------------------------------------------------
__global__ void k_acc(const float* __restrict__ rnn,
                      const float* __restrict__ wbuf,
                      float* __restrict__ part) {
    const int b = blockIdx.x;
    const int s = blockIdx.y;
    const int r = threadIdx.x * 4;
    const int lchunk = L_ / S_;                         // 128
    const float* wb = wbuf + (size_t)b * L_ + (size_t)s * lchunk;
    const float* rb = rnn + ((size_t)b * L_ + (size_t)s * lchunk) * R_ + r;
    f4 acc = {};
#pragma unroll 4
    for (int l = 0; l < lchunk; ++l) {
        const float wl = wb[l];
        f4 v = *(const f4*)rb;
        acc += wl * v;
        rb += R_;
    }
    *(f4*)(part + ((size_t)s * B_ + b) * R_ + r) = acc;
}

// ---------------------------------------------------------------------------
// Kernel 5: out[b,r] = scale[b] * sum_s partial[s][b][r]
// ---------------------------------------------------------------------------
__global__ void k_final(const float* __restrict__ part,
                        const float* __restrict__ scale,
                        float* __restrict__ out) {
    const int b = blockIdx.x;
    const int r = threadIdx.x * 4;
    f4 acc = {};
#pragma unroll
    for (int s = 0; s < S_; ++s)
        acc += *(const f4*)(part + ((size_t)s * B_ + b) * R_ + r);
    const float sc = scale[b];
    acc *= sc;
    *(f4*)(out + (size_t)b * R_ + r) = acc;
}

// ---------------------------------------------------------------------------
// Launch: 5 deterministic kernels, no atomics, no host-side sync, all on
// `stream` (graph-capture safe). Workspace fully rewritten each call.
// ---------------------------------------------------------------------------
extern "C" void kernel_launch(void* const* d_in, const int* in_sizes, int n_in,
                              void* d_out, int out_size, void* d_ws, size_t ws_size,
                              hipStream_t stream) {
    const float*     embs = (const float*)d_in[0];     // [B,L,E] f32
    const long long* prnt = (const long long*)d_in[1]; // [B,L]   i64
    const float*     lba  = (const float*)d_in[2];     // [B,L,R] f32
    const float*     rnn  = (const float*)d_in[3];     // [B,L,R] f32
    const float*     W    = (const float*)d_in[4];     // [E+R,R] f32
    float* out = (float*)d_out;                        // [B,R]   f32

    float* ws    = (float*)d_ws;
    float* wsum  = ws + WS_WSUM;
    float* wbuf  = ws + WS_W;
    float* scale = ws + WS_SCALE;
    float* part  = ws + WS_PART;

    k_wsum  <<<F_,                256, 0, stream>>>(W, wsum);
    k_scores<<<(B_ * L_) / 128,   256, 0, stream>>>(embs, prnt, lba, wsum, wbuf);
    k_scale <<<B_,                256, 0, stream>>>(wbuf, scale);
    k_acc   <<<dim3(B_, S_),      256, 0, stream>>>(rnn, wbuf, part);
    k_final <<<B_,                256, 0, stream>>>(part, scale, out);
}